// Initial_layer_79817672228969
// MI455X (gfx1250) — compile-verified
//
#include <hip/hip_runtime.h>

// Problem constants from the reference: G=14, B=64, T=2048, H=6.
// Each (b,t) point is independent: one thread per point, full state in VGPRs.

namespace {

constexpr int   G_  = 14;
constexpr int   H_  = 6;
constexpr int   BT  = 64 * 2048;                 // 131072 points
constexpr float LN2      = 0.69314718055994530942f;
constexpr float LOG_2PI  = 1.83787706640934548356f;  // ln(2*pi)

// One Gaussian product, in-place. a* get the (c3,c4) roles (normalizers come
// from a1[k], a2[k]); v* get the (n3,n4) roles; returns logC.
// Matches _gaussian_product semantics including _dnn's divide-by-zero guard.
__device__ __forceinline__ float gprod(float* a1, float* a2,
                                       float* v1, float* v2,
                                       float& b1, float& b2, int k) {
  const float C1 = a1[k];
  const float C2 = a2[k];

  const float r1  = __builtin_amdgcn_rcpf(C1);          // v_rcp_f32
  const float r2  = __builtin_amdgcn_rcpf(C2);
  const float r1s = (C1 == 0.0f) ? 0.0f : r1;           // _dnn guard
  const float r2s = (C2 == 0.0f) ? 0.0f : r2;

  const float var1 = r1 * r1;                           // 1/C1^2 (inf if C1==0, as in ref)
  const float var2 = r2 * r2;
  const float var3 = var1 + var2;

  const float rstd3 = __builtin_amdgcn_rsqf(var3);      // 1/sqrt(var3) for the divides
  const float std3  = __builtin_amdgcn_sqrtf(var3);     // sqrt(var3) for logC (keeps inf case right)
  const float var4  = var1 * var2 * __builtin_amdgcn_rcpf(var3);
  const float std4  = __builtin_amdgcn_sqrtf(var4);
  const float rden  = __builtin_amdgcn_rcpf(var3 * std4);

  const float logC =
      -LN2 * __builtin_amdgcn_logf(__builtin_fabsf(C1 * C2 * std4 * std3));

#pragma unroll
  for (int h = 0; h < H_; ++h) {
    const float cc1 = a1[h] * r1s;
    const float cc2 = a2[h] * r2s;
    const float nn1 = v1[h] * r1s;
    const float nn2 = v2[h] * r2s;
    a1[h] = (cc1 - cc2) * rstd3;                                   // c3
    a2[h] = __builtin_fmaf(cc1, var2, cc2 * var1) * rden;          // c4
    v1[h] = (nn1 - nn2) * rstd3;                                   // n3
    v2[h] = __builtin_fmaf(nn1, var2, nn2 * var1) * rden;          // n4
  }
  const float bb1 = b1 * r1s;
  const float bb2 = b2 * r2s;
  b1 = (bb1 - bb2) * rstd3;                                        // b3
  b2 = __builtin_fmaf(bb1, var2, bb2 * var1) * rden;               // b4
  return logC;
}

} // namespace

__global__ __launch_bounds__(256, 1) void gaussian_chain_kernel(
    const float* __restrict__ cur,   // (G,B,T,H) f32
    const float* __restrict__ nxt,   // (G,B,T,H) f32
    const float* __restrict__ bias,  // (G,B,T)   f32
    float* __restrict__ out) {       // kept_n (6,B,T,H) ++ kept_b (6,B,T) ++ LC (B,T)
  const int p = blockIdx.x * blockDim.x + threadIdx.x;   // BT is a multiple of 256
  if (p >= BT) return;

  float c[G_][H_];
  float v[G_][H_];
  float b[G_];

  // Load: element (g,p) is 6 contiguous floats at offset (g*BT+p)*6 -> 8B aligned.
#pragma unroll
  for (int g = 0; g < G_; ++g) {
    const float2* pc = reinterpret_cast<const float2*>(cur) + ((size_t)g * BT + p) * 3;
    const float2* pn = reinterpret_cast<const float2*>(nxt) + ((size_t)g * BT + p) * 3;
#pragma unroll
    for (int j = 0; j < 3; ++j) {
      const float2 tc = pc[j];
      const float2 tn = pn[j];
      c[g][2 * j]     = tc.x;
      c[g][2 * j + 1] = tc.y;
      v[g][2 * j]     = tn.x;
      v[g][2 * j + 1] = tn.y;
    }
    b[g] = bias[(size_t)g * BT + p];
  }

  float LC = 0.0f;

  // Phase 1: normalize by cur; list shrinks 14 -> 8.
#pragma unroll
  for (int k = 0; k < H_; ++k) {
    const int n = G_ - k;
#pragma unroll
    for (int i = 0; i < n - 1; ++i)
      LC += gprod(c[i], c[i + 1], v[i], v[i + 1], b[i], b[i + 1], k);
    LC -= LN2 * __builtin_amdgcn_logf(__builtin_fabsf(c[n - 1][k]));
    // pop: simply stop using index n-1 from here on
  }

  // Phase 2: normalize by nxt; list shrinks 8 -> 2; popped tails are outputs.
#pragma unroll
  for (int k = 0; k < H_; ++k) {
    const int n = (G_ - H_) - k;   // 8 - k
#pragma unroll
    for (int i = 0; i < n - 1; ++i)
      LC += gprod(v[i], v[i + 1], c[i], c[i + 1], b[i], b[i + 1], k);

    // kept_n[k] = nxt_l[n-1], kept_b[k] = b_l[n-1]; store now to free registers.
    float2* po = reinterpret_cast<float2*>(out) + ((size_t)k * BT + p) * 3;
    po[0] = make_float2(v[n - 1][0], v[n - 1][1]);
    po[1] = make_float2(v[n - 1][2], v[n - 1][3]);
    po[2] = make_float2(v[n - 1][4], v[n - 1][5]);
    out[(size_t)H_ * BT * H_ + (size_t)k * BT + p] = b[n - 1];
  }

  // Remaining two biases feed norm_log_gaussian.
  LC += -0.5f * (LOG_2PI + b[0] * b[0]) - 0.5f * (LOG_2PI + b[1] * b[1]);
  out[(size_t)H_ * BT * H_ + (size_t)H_ * BT + p] = LC;
}

extern "C" void kernel_launch(void* const* d_in, const int* in_sizes, int n_in,
                              void* d_out, int out_size, void* d_ws, size_t ws_size,
                              hipStream_t stream) {
  (void)in_sizes; (void)n_in; (void)d_ws; (void)ws_size; (void)out_size;
  const float* cur  = (const float*)d_in[0];
  const float* nxt  = (const float*)d_in[1];
  const float* bias = (const float*)d_in[2];
  float* out = (float*)d_out;

  const int threads = 256;                 // 8 wave32s per workgroup
  const int blocks  = (BT + threads - 1) / threads;   // 512
  gaussian_chain_kernel<<<blocks, threads, 0, stream>>>(cur, nxt, bias, out);
}